// KuramotoSivashinsky_76347338654370
// MI455X (gfx1250) — compile-verified
//
#include <hip/hip_runtime.h>
#include <hip/hip_bf16.h>
#include <math.h>

typedef __attribute__((ext_vector_type(2))) float v2f;
typedef __attribute__((ext_vector_type(8))) float v8f;

#define NPTS 64
#define ROWS_PER_BLOCK 128

// ---------------------------------------------------------------------------
// Setup: build the two circulant operator matrices, packed for WMMA B-frags.
// B-matrix element (k, n) of the GEMM  C = U * B  lives at packed index
//   (k>>1)*128 + n*2 + (k&1)
// so a lane's (K, K+1) pair at column n is one contiguous 8-byte LDS load.
//   BD[k][n] = D[n][k] = -(1/64) sum_j k_j sin(2*pi*j*(n-k)/64)
//   BL[k][n] = L[n][k] =  (1/64) sum_j (k_j^2 - k_j^4) cos(2*pi*j*(n-k)/64)
// with k_j = (pi/16) * ((j<32) ? j : j-64)   [L=32, N=64 -> d=0.5]
// ---------------------------------------------------------------------------
__global__ void ks_setup_mats(float* __restrict__ mats) {
  int tid = blockIdx.x * blockDim.x + threadIdx.x;   // 0..4095
  if (tid >= NPTS * NPTS) return;
  int k = tid >> 6;
  int n = tid & 63;
  float sd = 0.0f, sl = 0.0f;
  const float PI = 3.14159265358979323846f;
  for (int j = 0; j < NPTS; ++j) {
    float fj  = (j < 32) ? (float)j : (float)(j - 64);
    float kj  = (PI / 16.0f) * fj;
    float ang = (2.0f * PI) * (float)j * (float)(n - k) * (1.0f / 64.0f);
    float s, c;
    __sincosf(ang, &s, &c);
    float k2 = kj * kj;
    sd += kj * s;
    sl += (k2 - k2 * k2) * c;
  }
  int kk = k >> 1, p = k & 1;
  int idx = kk * 128 + n * 2 + p;
  mats[idx]             = -sd * (1.0f / 64.0f);   // derivative operator
  mats[NPTS * NPTS + idx] =  sl * (1.0f / 64.0f); // (k^2 - k^4) operator
}

// ---------------------------------------------------------------------------
// Main: out[b,:] = -0.5 * u[b,:] * (D u[b,:]) + (L u[b,:])
// 256 threads = 8 waves; each wave owns 16 rows; block owns 128 rows.
// ---------------------------------------------------------------------------
__global__ __launch_bounds__(256) void ks_main(const float* __restrict__ u,
                                               float* __restrict__ out,
                                               const float* __restrict__ mats,
                                               int nrows) {
  __shared__ __align__(16) float sBD[NPTS * NPTS];   // 16 KB
  __shared__ __align__(16) float sBL[NPTS * NPTS];   // 16 KB
  __shared__ __align__(16) float sU[ROWS_PER_BLOCK * NPTS]; // 32 KB

  const int tid = threadIdx.x;

  // Stage operator matrices (coalesced float4).
  const float4* m4  = (const float4*)mats;
  float4* sbd4 = (float4*)sBD;
  float4* sbl4 = (float4*)sBL;
#pragma unroll
  for (int i = 0; i < 4; ++i) {
    sbd4[tid + 256 * i] = m4[tid + 256 * i];
    sbl4[tid + 256 * i] = m4[1024 + tid + 256 * i];
  }

  // Stage 128x64 u tile (coalesced float4): 2048 float4 / 256 threads = 8.
  const long rowBase = (long)blockIdx.x * ROWS_PER_BLOCK;
  const float4* u4 = (const float4*)(u + rowBase * NPTS);
  float4* su4 = (float4*)sU;
#pragma unroll
  for (int i = 0; i < 8; ++i) su4[tid + 256 * i] = u4[tid + 256 * i];

  __syncthreads();

  const int wave = tid >> 5;
  const int lane = tid & 31;
  const int r = lane & 15;   // row-in-tile (A) / column-in-tile (B,C)
  const int h = lane >> 4;   // half-wave selector
  const int waveRow = wave * 16;

  // A-fragments: 16x4 fp32 per K-step. Lane layout (ISA 7.12.2):
  //   lanes 0-15 : M=r, VGPR0=K0, VGPR1=K1 ; lanes 16-31 : M=r, K2/K3.
  v2f afr[16];
  const float* uRow = &sU[(waveRow + r) * NPTS];
#pragma unroll
  for (int ks = 0; ks < 16; ++ks)
    afr[ks] = *(const v2f*)(uRow + 4 * ks + 2 * h);

#pragma unroll
  for (int nt = 0; nt < 4; ++nt) {
    v8f accD = {};
    v8f accL = {};
    const int ncol = nt * 16 + r;
#pragma unroll
    for (int ks = 0; ks < 16; ++ks) {
      const int kk = 2 * ks + h;                       // packed K-pair index
      v2f bd = *(const v2f*)(&sBD[kk * 128 + ncol * 2]);
      v2f bl = *(const v2f*)(&sBL[kk * 128 + ncol * 2]);
      accD = __builtin_amdgcn_wmma_f32_16x16x4_f32(
          false, afr[ks], false, bd, (short)0, accD, false, false);
      accL = __builtin_amdgcn_wmma_f32_16x16x4_f32(
          false, afr[ks], false, bl, (short)0, accL, false, false);
    }
    // Epilogue: C-layout VGPR v holds (M = v + 8*h, N = r) of this tile.
#pragma unroll
    for (int v = 0; v < 8; ++v) {
      const int mrow = waveRow + v + 8 * h;
      const float uval = sU[mrow * NPTS + nt * 16 + r];
      const float res = -0.5f * uval * accD[v] + accL[v];
      out[(rowBase + mrow) * NPTS + nt * 16 + r] = res;
    }
  }
}

extern "C" void kernel_launch(void* const* d_in, const int* in_sizes, int n_in,
                              void* d_out, int out_size, void* d_ws, size_t ws_size,
                              hipStream_t stream) {
  // d_in[0] = t (unused scalar), d_in[1] = u [B,64] float32
  const float* u = (const float*)d_in[1];
  float* out = (float*)d_out;
  float* mats = (float*)d_ws;                 // 2 * 64*64 floats = 32 KB

  const int nrows = in_sizes[1] / NPTS;       // 1048576

  ks_setup_mats<<<(NPTS * NPTS + 255) / 256, 256, 0, stream>>>(mats);

  const int grid = nrows / ROWS_PER_BLOCK;    // 8192
  ks_main<<<grid, 256, 0, stream>>>(u, out, mats, nrows);
}